// GCNEncoder_18519898980762
// MI455X (gfx1250) — compile-verified
//
#include <hip/hip_runtime.h>
#include <hip/hip_bf16.h>

// Problem constants (match reference)
#define NN   50000
#define EE   800000
#define GG   500
#define DD   64
#define BN_EPS 1e-5f

typedef float v2f __attribute__((ext_vector_type(2)));
typedef float v8f __attribute__((ext_vector_type(8)));

// ---------------------------------------------------------------------------
// Degree / normalization
// ---------------------------------------------------------------------------
__global__ void k_deg_init(float* __restrict__ deg) {
    int i = blockIdx.x * blockDim.x + threadIdx.x;
    if (i < NN) deg[i] = 1.0f;               // self-loop contributes 1
}

__global__ void k_deg_accum(const int* __restrict__ ei, float* __restrict__ deg) {
    int e = blockIdx.x * blockDim.x + threadIdx.x;
    if (e < EE) atomicAdd(&deg[ei[EE + e]], 1.0f);   // dst row
}

__global__ void k_deg_rsqrt(float* __restrict__ deg) {
    int i = blockIdx.x * blockDim.x + threadIdx.x;
    if (i < NN) deg[i] = rsqrtf(deg[i]);     // dinv
}

// ---------------------------------------------------------------------------
// GEMM: H[N,64] = X[N,64] @ W[64,64] using V_WMMA_F32_16X16X4_F32
// One wave -> one 16x16 tile; 4 waves/block cover 16 rows x 64 cols.
// Grid = N/16 = 3125 blocks (exact, no tails -> EXEC all-1s for WMMA).
// ---------------------------------------------------------------------------
__global__ void k_gemm_wmma(const float* __restrict__ X,
                            const float* __restrict__ W,
                            float* __restrict__ H) {
    const int lane  = threadIdx.x & 31;
    const int wave  = threadIdx.x >> 5;        // 0..3 -> 16-col tile
    const int row0  = blockIdx.x * 16;
    const int col0  = wave * 16;

    const int m     = lane & 15;               // M index within tile
    const int khalf = (lane >> 4) << 1;        // 0 (lanes 0-15) or 2 (lanes 16-31)
    const int n     = lane & 15;               // N index within tile

    const float* xrow = X + (size_t)(row0 + m) * DD;

    v8f c = {};
#pragma unroll
    for (int kk = 0; kk < 16; ++kk) {          // K = 64 = 16 * 4
        const int k = kk * 4 + khalf;
        v2f a, b;
        // A (16x4 f32): v0 = K=khalf, v1 = K=khalf+1, per ISA layout
        a.x = xrow[k];
        a.y = xrow[k + 1];
        // B (4x16 f32): same K split across lane halves, N striped over lanes
        b.x = W[(size_t)k * DD + col0 + n];
        b.y = W[(size_t)(k + 1) * DD + col0 + n];
        c = __builtin_amdgcn_wmma_f32_16x16x4_f32(
                /*neg_a=*/false, a, /*neg_b=*/false, b,
                /*c_mod=*/(short)0, c, /*reuse_a=*/false, /*reuse_b=*/false);
    }

    // C/D layout: VGPR r -> M = r + 8*(lane>=16), N = lane&15
    const int mrow = row0 + ((lane >> 4) << 3);
#pragma unroll
    for (int r = 0; r < 8; ++r)
        H[(size_t)(mrow + r) * DD + col0 + n] = c[r];
}

// ---------------------------------------------------------------------------
// Self-loop term + bias: A = H * dinv^2 + b  (initializes A fully each call)
// ---------------------------------------------------------------------------
__global__ void k_selfloop_init(const float* __restrict__ H,
                                const float* __restrict__ dinv,
                                const float* __restrict__ bias,
                                float* __restrict__ A) {
    int i = blockIdx.x * blockDim.x + threadIdx.x;
    if (i >= NN * DD) return;
    int node = i >> 6, f = i & 63;
    float di = dinv[node];
    A[i] = H[i] * di * di + bias[f];
}

// ---------------------------------------------------------------------------
// Edge scatter: A[dst] += dinv[src]*dinv[dst] * H[src], 4 features/thread
// ---------------------------------------------------------------------------
__global__ void k_edge_scatter(const float* __restrict__ H,
                               const int* __restrict__ ei,
                               const float* __restrict__ dinv,
                               float* __restrict__ A) {
    long long t = (long long)blockIdx.x * blockDim.x + threadIdx.x;
    int e = (int)(t >> 4);
    if (e >= EE) return;
    int f4 = ((int)t & 15) * 4;
    int s = ei[e];
    int d = ei[EE + e];
    float norm = dinv[s] * dinv[d];
    const float4 hv = *(const float4*)(H + (size_t)s * DD + f4);
    float* ap = A + (size_t)d * DD + f4;
    atomicAdd(ap + 0, norm * hv.x);
    atomicAdd(ap + 1, norm * hv.y);
    atomicAdd(ap + 2, norm * hv.z);
    atomicAdd(ap + 3, norm * hv.w);
}

// ---------------------------------------------------------------------------
// BatchNorm stats: one block per feature column; writes scale/shift
// ---------------------------------------------------------------------------
__global__ void k_bn_stats(const float* __restrict__ A,
                           const float* __restrict__ gamma,
                           const float* __restrict__ beta,
                           float* __restrict__ ss) {
    __shared__ float s_sum[256];
    __shared__ float s_sq[256];
    const int f = blockIdx.x;                 // 0..63
    float sum = 0.f, sq = 0.f;
    for (int node = threadIdx.x; node < NN; node += blockDim.x) {
        float v = A[(size_t)node * DD + f];
        sum += v;
        sq  += v * v;
    }
    s_sum[threadIdx.x] = sum;
    s_sq[threadIdx.x]  = sq;
    __syncthreads();
    for (int s = 128; s > 0; s >>= 1) {
        if (threadIdx.x < s) {
            s_sum[threadIdx.x] += s_sum[threadIdx.x + s];
            s_sq[threadIdx.x]  += s_sq[threadIdx.x + s];
        }
        __syncthreads();
    }
    if (threadIdx.x == 0) {
        float mean = s_sum[0] * (1.0f / NN);
        float var  = s_sq[0] * (1.0f / NN) - mean * mean;
        float inv  = rsqrtf(var + BN_EPS);
        float sc   = gamma[f] * inv;
        ss[f]      = sc;                       // scale
        ss[DD + f] = beta[f] - mean * sc;      // shift
    }
}

__global__ void k_bn_apply(float* __restrict__ A,
                           const float* __restrict__ ss, int do_relu) {
    int i = blockIdx.x * blockDim.x + threadIdx.x;
    if (i >= NN * DD) return;
    int f = i & 63;
    float v = fmaf(A[i], ss[f], ss[DD + f]);
    if (do_relu) v = fmaxf(v, 0.0f);
    A[i] = v;
}

// ---------------------------------------------------------------------------
// Global mean pool
// ---------------------------------------------------------------------------
__global__ void k_pool_init(float* __restrict__ out, float* __restrict__ cnt) {
    int i = blockIdx.x * blockDim.x + threadIdx.x;
    if (i < GG * DD) out[i] = 0.0f;
    if (i < GG) cnt[i] = 0.0f;
}

__global__ void k_pool_accum(const float* __restrict__ A,
                             const int* __restrict__ batch,
                             float* __restrict__ out,
                             float* __restrict__ cnt) {
    long long t = (long long)blockIdx.x * blockDim.x + threadIdx.x;
    int node = (int)(t >> 4);
    if (node >= NN) return;
    int f4 = ((int)t & 15) * 4;
    int g = batch[node];
    const float4 v = *(const float4*)(A + (size_t)node * DD + f4);
    float* op = out + (size_t)g * DD + f4;
    atomicAdd(op + 0, v.x);
    atomicAdd(op + 1, v.y);
    atomicAdd(op + 2, v.z);
    atomicAdd(op + 3, v.w);
    if ((t & 15) == 0) atomicAdd(&cnt[g], 1.0f);
}

__global__ void k_pool_div(float* __restrict__ out, const float* __restrict__ cnt) {
    int i = blockIdx.x * blockDim.x + threadIdx.x;
    if (i >= GG * DD) return;
    out[i] /= fmaxf(cnt[i >> 6], 1.0f);
}

// ---------------------------------------------------------------------------
// Launch
// ---------------------------------------------------------------------------
extern "C" void kernel_launch(void* const* d_in, const int* in_sizes, int n_in,
                              void* d_out, int out_size, void* d_ws, size_t ws_size,
                              hipStream_t stream) {
    const float* x     = (const float*)d_in[0];
    const int*   ei    = (const int*)d_in[1];     // [2,E]: row0=src, row1=dst
    const int*   batch = (const int*)d_in[2];
    const float* Wm[3]  = { (const float*)d_in[3],  (const float*)d_in[7],  (const float*)d_in[11] };
    const float* bv[3]  = { (const float*)d_in[4],  (const float*)d_in[8],  (const float*)d_in[12] };
    const float* gm[3]  = { (const float*)d_in[5],  (const float*)d_in[9],  (const float*)d_in[13] };
    const float* bt[3]  = { (const float*)d_in[6],  (const float*)d_in[10], (const float*)d_in[14] };

    float* ws   = (float*)d_ws;
    float* dinv = ws;                                    // N
    float* h    = ws + NN;                               // N*64
    float* a    = ws + NN + (size_t)NN * DD;             // N*64
    float* ss   = ws + NN + (size_t)2 * NN * DD;         // 128 (scale|shift)
    float* cnt  = ss + 2 * DD;                           // G
    float* out  = (float*)d_out;                         // G*64

    const int T = 256;
    // gcn_norm
    k_deg_init  <<<(NN + T - 1) / T, T, 0, stream>>>(dinv);
    k_deg_accum <<<(EE + T - 1) / T, T, 0, stream>>>(ei, dinv);
    k_deg_rsqrt <<<(NN + T - 1) / T, T, 0, stream>>>(dinv);

    const float* cur = x;
    for (int l = 0; l < 3; ++l) {
        k_gemm_wmma    <<<NN / 16, 128, 0, stream>>>(cur, Wm[l], h);
        k_selfloop_init<<<(NN * DD + T - 1) / T, T, 0, stream>>>(h, dinv, bv[l], a);
        k_edge_scatter <<<((long long)EE * 16 + T - 1) / T, T, 0, stream>>>(h, ei, dinv, a);
        k_bn_stats     <<<DD, 256, 0, stream>>>(a, gm[l], bt[l], ss);
        k_bn_apply     <<<(NN * DD + T - 1) / T, T, 0, stream>>>(a, ss, l < 2 ? 1 : 0);
        cur = a;
    }

    k_pool_init <<<(GG * DD + T - 1) / T, T, 0, stream>>>(out, cnt);
    k_pool_accum<<<((long long)NN * 16 + T - 1) / T, T, 0, stream>>>(a, batch, out, cnt);
    k_pool_div  <<<(GG * DD + T - 1) / T, T, 0, stream>>>(out, cnt);
}